// GMNLayer_84112639525110
// MI455X (gfx1250) — compile-verified
//
#include <hip/hip_runtime.h>
#include <hip/hip_bf16.h>

#define NN    1024      // nodes
#define EDG   65536     // edges
#define NE    16
#define NINP  64

typedef float v2f __attribute__((ext_vector_type(2)));
typedef float v8f __attribute__((ext_vector_type(8)));

// ---------------------------------------------------------------------------
// Generic f32 WMMA GEMM:  C[M x N] = act(alpha * A[M x K] @ B[K x N])
// Row-major, lda = K, ldb = N, ldc parameterized. M,N multiples of 64,
// K multiple of 32. Block = 128 threads (4 waves); block tile 64x64; wave w
// owns rows [64*bx + 16w, +16) x 64 cols (four 16x16 f32 accumulators).
//
// LDS staging is double-buffered; B is staged pre-swizzled in WMMA fragment
// order (k-pairs interleaved) so every fragment is one aligned ds_load_b64:
//   lsB[buf][(k>>1)*128 + n*2 + (k&1)] = B[k0+k, n0+n]
// A tile is plain row-major 64x32 (k-pair contiguous -> ds_load_b64).
// ---------------------------------------------------------------------------
template <bool RELU>
__global__ void wmma_gemm_f32(const float* __restrict__ A,
                              const float* __restrict__ B,
                              float* __restrict__ C,
                              int N, int K, int ldc, float alpha)
{
    __shared__ float lsA[2][64 * 32];
    __shared__ float lsB[2][32 * 64];

    const int tid  = threadIdx.x;      // 0..127
    const int wave = tid >> 5;         // 0..3
    const int lane = tid & 31;
    const int m0   = blockIdx.x * 64;
    const int n0   = blockIdx.y * 64;

    const int mlane = lane & 15;           // M (A frag) / N (B frag) index
    const int khalf = (lane >> 4) << 1;    // 0 or 2: per-lane K sub-slot

    v8f acc[4] = {};

    // stage one 64x32 A tile + swizzled 32x64 B tile into buffer `buf`
    auto stage = [&](int k0, int buf) {
        for (int i = tid; i < 64 * 8; i += 128) {          // 512 x float4
            const int r = i >> 3, c = (i & 7) << 2;
            const float4 v = *(const float4*)&A[(size_t)(m0 + r) * K + k0 + c];
            *(float4*)&lsA[buf][r * 32 + c] = v;
        }
        for (int i = tid; i < 32 * 16; i += 128) {         // 512 x float4
            const int r = i >> 4, c = (i & 15) << 2;
            const float4 v = *(const float4*)&B[(size_t)(k0 + r) * N + n0 + c];
            float* dst = &lsB[buf][(r >> 1) * 128 + (r & 1)];
            dst[(c + 0) * 2] = v.x;
            dst[(c + 1) * 2] = v.y;
            dst[(c + 2) * 2] = v.z;
            dst[(c + 3) * 2] = v.w;
        }
    };

    const int nchunks = K >> 5;
    stage(0, 0);
    __syncthreads();

    for (int ch = 0; ch < nchunks; ++ch) {
        const int cur = ch & 1;
        if (ch + 1 < nchunks) stage((ch + 1) << 5, cur ^ 1);  // prefetch next

        const float* __restrict__ pa = &lsA[cur][(wave * 16 + mlane) * 32];
        const float* __restrict__ pb = &lsB[cur][mlane * 2];
#pragma unroll
        for (int ks = 0; ks < 8; ++ks) {        // 8 x (K=4) = 32
            const int kk = ks * 4 + khalf;      // even -> 8B-aligned pair
            const v2f a = *(const v2f*)&pa[kk];
            const int prow = (kk >> 1) * 128;   // swizzled pair-row base
#pragma unroll
            for (int f = 0; f < 4; ++f) {
                const v2f b = *(const v2f*)&pb[prow + f * 32];
                acc[f] = __builtin_amdgcn_wmma_f32_16x16x4_f32(
                    /*neg_a=*/false, a, /*neg_b=*/false, b,
                    /*c_mod=*/(short)0, acc[f],
                    /*reuse_a=*/false, /*reuse_b=*/false);
            }
        }
        __syncthreads();   // reads of cur done; writes to cur^1 visible
    }

    // C/D layout: VGPR v, lane t -> row = v + 8*(t>=16), col = t&15
    const int rbase = m0 + wave * 16 + ((lane >> 4) << 3);
#pragma unroll
    for (int f = 0; f < 4; ++f) {
        const int col = n0 + f * 16 + mlane;
#pragma unroll
        for (int v = 0; v < 8; ++v) {
            float val = acc[f][v] * alpha;
            if (RELU) val = fmaxf(val, 0.f);
            C[(size_t)(rbase + v) * ldc + col] = val;
        }
    }
}

// ---------------------------------------------------------------------------
// h_node = relu(x @ fcnode1 / 128) @ fcnode2 / 16          [1024 x 16]
// ---------------------------------------------------------------------------
__global__ void node_mlp_kernel(const float* __restrict__ x,
                                const float* __restrict__ w1,  // [64,128]
                                const float* __restrict__ w2,  // [128,16]
                                float* __restrict__ h_node)
{
    __shared__ float hid[128];
    const int n = blockIdx.x;
    const int j = threadIdx.x;           // 0..127
    float acc = 0.f;
    for (int d = 0; d < 64; ++d) acc += x[n * 64 + d] * w1[d * 128 + j];
    hid[j] = fmaxf(acc * (1.f / 128.f), 0.f);
    __syncthreads();
    if (j < 16) {
        float o = 0.f;
        for (int k = 0; k < 128; ++k) o += hid[k] * w2[k * 16 + j];
        h_node[n * 16 + j] = o * (1.f / 16.f);
    }
}

// ---------------------------------------------------------------------------
// Four edge MLPs in one pass: A = mlp41(SP), B = mlp51(SP),
// C2 = mlp21(SP) * mlp31(SP).   mlp(sp) = (relu(sp@w1/16)/32) @ w2
// ---------------------------------------------------------------------------
__device__ __forceinline__ void edge_mlp(const float sp[16],
                                         const float* __restrict__ w1s,
                                         const float* __restrict__ w2s,
                                         float out[16])
{
    float hid[32];
#pragma unroll
    for (int j = 0; j < 32; ++j) {
        float a = 0.f;
#pragma unroll
        for (int d = 0; d < 16; ++d) a += sp[d] * w1s[d * 32 + j];
        hid[j] = fmaxf(a * (1.f / 16.f), 0.f) * (1.f / 32.f);
    }
#pragma unroll
    for (int o = 0; o < 16; ++o) {
        float a = 0.f;
#pragma unroll
        for (int j = 0; j < 32; ++j) a += hid[j] * w2s[j * 16 + o];
        out[o] = a;
    }
}

__global__ void edge_mlps_kernel(const float* __restrict__ SP,
                                 const float* w21, const float* w22,
                                 const float* w31, const float* w32,
                                 const float* w41, const float* w42,
                                 const float* w51, const float* w52,
                                 float* __restrict__ Aq, float* __restrict__ Bq,
                                 float* __restrict__ C2)
{
    __shared__ float lw[4][1024];      // 4 MLP pairs x (512 w1 + 512 w2)
    const int tid = threadIdx.x;
    const float* srcs[8] = {w21, w22, w31, w32, w41, w42, w51, w52};
    for (int i = tid; i < 4096; i += 256) {
        int pair = i >> 10, off = i & 1023;
        lw[pair][off] = (off < 512) ? srcs[pair * 2][off]
                                    : srcs[pair * 2 + 1][off - 512];
    }
    __syncthreads();
    const int e = blockIdx.x * 256 + tid;
    float sp[16];
#pragma unroll
    for (int i = 0; i < 16; ++i) sp[i] = SP[e * 16 + i];
    float m2[16], m3[16], m4[16], m5[16];
    edge_mlp(sp, &lw[0][0], &lw[0][512], m2);
    edge_mlp(sp, &lw[1][0], &lw[1][512], m3);
    edge_mlp(sp, &lw[2][0], &lw[2][512], m4);
    edge_mlp(sp, &lw[3][0], &lw[3][512], m5);
#pragma unroll
    for (int i = 0; i < 16; ++i) {
        C2[e * 16 + i] = m2[i] * m3[i];
        Aq[e * 16 + i] = m4[i];
        Bq[e * 16 + i] = m5[i];
    }
}

// Scatter channel c of A,B into dense Xm, Ym (duplicates accumulate)
__global__ void scatter_ab_kernel(const int* __restrict__ ei,
                                  const float* __restrict__ Aq,
                                  const float* __restrict__ Bq,
                                  float* __restrict__ Xm, float* __restrict__ Ym,
                                  int c)
{
    const int e = blockIdx.x * 256 + threadIdx.x;
    const int i = ei[e], k = ei[EDG + e];
    atomicAdd(&Xm[(size_t)i * NN + k], Aq[e * 16 + c]);
    atomicAdd(&Ym[(size_t)i * NN + k], Bq[e * 16 + c]);
}

__global__ void gather_prod_kernel(const int* __restrict__ ei,
                                   const float* __restrict__ prod,
                                   float* __restrict__ tmpm, int c)
{
    const int e = blockIdx.x * 256 + threadIdx.x;
    const int i = ei[e], k = ei[EDG + e];
    tmpm[e * 16 + c] = prod[(size_t)i * NN + k];
}

// tmp = concat([SP, C2, tmp_matmul, tmp_diag], axis=1)   [E x 64]
__global__ void build_tmp_kernel(const int* __restrict__ ei,
                                 const float* __restrict__ SP,
                                 const float* __restrict__ C2,
                                 const float* __restrict__ tmpm,
                                 const float* __restrict__ h_node,
                                 float* __restrict__ tmp)
{
    const int e = blockIdx.x * 256 + threadIdx.x;
    const int s = ei[e], t = ei[EDG + e];
    const bool diag = (s == t);
#pragma unroll
    for (int i = 0; i < 16; ++i) {
        tmp[(size_t)e * 64 + i]      = SP[e * 16 + i];
        tmp[(size_t)e * 64 + 16 + i] = C2[e * 16 + i];
        tmp[(size_t)e * 64 + 32 + i] = tmpm[e * 16 + i];
        tmp[(size_t)e * 64 + 48 + i] = diag ? h_node[s * 16 + i] : 0.f;
    }
}

// segment_sum: agg[dst, k*64+d] += edge_attr[e,k] * x[src,d]
__global__ void agg_scatter_kernel(const int* __restrict__ ei,
                                   const float* __restrict__ ea, // [E,64]
                                   const float* __restrict__ x,
                                   float* __restrict__ agg)      // [1024,1024]
{
    const int tid = threadIdx.x;
    const int e = blockIdx.x * 4 + (tid >> 6);
    const int d = tid & 63;
    const int s = ei[e], t = ei[EDG + e];
    const float xv = x[s * 64 + d];
#pragma unroll
    for (int k = 0; k < 16; ++k)
        atomicAdd(&agg[(size_t)t * 1024 + k * 64 + d], ea[(size_t)e * 64 + k] * xv);
}

// gate = ((x@fc11 + b11)/64) * ((x@fc12 + b12)/64) -> x_out cols 64..127
__global__ void gate_kernel(const float* __restrict__ x,
                            const float* __restrict__ w11, const float* __restrict__ b11,
                            const float* __restrict__ w12, const float* __restrict__ b12,
                            float* __restrict__ out)
{
    const int n = blockIdx.x, o = threadIdx.x;
    float g1 = b11[o], g2 = b12[o];
    for (int d = 0; d < 64; ++d) {
        const float xv = x[n * 64 + d];
        g1 += xv * w11[d * 64 + o];
        g2 += xv * w12[d * 64 + o];
    }
    out[(size_t)n * 128 + 64 + o] = (g1 * (1.f / 64.f)) * (g2 * (1.f / 64.f));
}

// ---------------------------------------------------------------------------
extern "C" void kernel_launch(void* const* d_in, const int* in_sizes, int n_in,
                              void* d_out, int out_size, void* d_ws, size_t ws_size,
                              hipStream_t stream)
{
    const float* x    = (const float*)d_in[0];
    const int*   ei   = (const int*)d_in[1];
    const float* SP   = (const float*)d_in[2];
    const float* w21  = (const float*)d_in[3];
    const float* w22  = (const float*)d_in[4];
    const float* w31  = (const float*)d_in[5];
    const float* w32  = (const float*)d_in[6];
    const float* w41  = (const float*)d_in[7];
    const float* w42  = (const float*)d_in[8];
    const float* w51  = (const float*)d_in[9];
    const float* w52  = (const float*)d_in[10];
    const float* w61  = (const float*)d_in[11];   // [64,128]
    const float* w62  = (const float*)d_in[12];   // [128,64]
    const float* wn1  = (const float*)d_in[13];   // [64,128]
    const float* wn2  = (const float*)d_in[14];   // [128,16]
    const float* cw   = (const float*)d_in[15];   // [16,64,64] -> [1024,64]
    const float* w11  = (const float*)d_in[16];
    const float* b11  = (const float*)d_in[17];
    const float* w12  = (const float*)d_in[18];
    const float* b12  = (const float*)d_in[19];

    float* out   = (float*)d_out;
    float* xout  = out;                     // [1024, 128]
    float* eaout = out + (size_t)NN * 128;  // [65536, 64]

    // workspace carve-up (floats); ~80 MB total
    float* ws     = (float*)d_ws;
    float* h_node = ws;  ws += (size_t)NN * NE;
    float* Aq     = ws;  ws += (size_t)EDG * NE;
    float* Bq     = ws;  ws += (size_t)EDG * NE;
    float* C2     = ws;  ws += (size_t)EDG * NE;
    float* tmpm   = ws;  ws += (size_t)EDG * NE;
    float* Xm     = ws;  ws += (size_t)NN * NN;     // Xm and Ym adjacent:
    float* Ym     = ws;  ws += (size_t)NN * NN;     //  one 8MB memset zeroes both
    float* prod   = ws;  ws += (size_t)NN * NN;
    float* tmp    = ws;  ws += (size_t)EDG * 64;
    float* hid    = ws;  ws += (size_t)EDG * 128;
    float* agg    = ws;  ws += (size_t)NN * 1024;

    // independent node-side work
    gate_kernel<<<NN, 64, 0, stream>>>(x, w11, b11, w12, b12, xout);
    node_mlp_kernel<<<NN, 128, 0, stream>>>(x, wn1, wn2, h_node);

    // edge MLPs -> A, B, C2
    edge_mlps_kernel<<<EDG / 256, 256, 0, stream>>>(SP, w21, w22, w31, w32,
                                                    w41, w42, w51, w52,
                                                    Aq, Bq, C2);

    // per-channel dense 1024^3 f32 WMMA GEMM (working set fits L2)
    for (int c = 0; c < NE; ++c) {
        hipMemsetAsync(Xm, 0, (size_t)NN * NN * 2 * sizeof(float), stream);
        scatter_ab_kernel<<<EDG / 256, 256, 0, stream>>>(ei, Aq, Bq, Xm, Ym, c);
        wmma_gemm_f32<false><<<dim3(NN / 64, NN / 64), 128, 0, stream>>>(
            Xm, Ym, prod, NN, NN, NN, 1.f);
        gather_prod_kernel<<<EDG / 256, 256, 0, stream>>>(ei, prod, tmpm, c);
    }

    // edge_attr = relu(relu(tmp @ w61) @ w62)
    build_tmp_kernel<<<EDG / 256, 256, 0, stream>>>(ei, SP, C2, tmpm, h_node, tmp);
    wmma_gemm_f32<true><<<dim3(EDG / 64, 128 / 64), 128, 0, stream>>>(
        tmp, w61, hid, 128, 64, 128, 1.f);
    wmma_gemm_f32<true><<<dim3(EDG / 64, 1), 128, 0, stream>>>(
        hid, w62, eaout, 64, 128, 64, 1.f);

    // SpectConv: segment-sum then contraction, scaled by 1/NOUT1, into x_out cols 0..63
    hipMemsetAsync(agg, 0, (size_t)NN * 1024 * sizeof(float), stream);
    agg_scatter_kernel<<<EDG / 4, 256, 0, stream>>>(ei, eaout, x, agg);
    wmma_gemm_f32<false><<<dim3(NN / 64, 1), 128, 0, stream>>>(
        agg, cw, xout, 64, 1024, 128, 1.f / 64.f);
}